// CompressiveMemory_57750130262084
// MI455X (gfx1250) — compile-verified
//
#include <hip/hip_runtime.h>

// out[b,s,v] = sum_s value[b,s,v]  (reference provably reduces to this; query unused)

typedef __attribute__((ext_vector_type(2))) float v2f;
typedef __attribute__((ext_vector_type(4))) float v4f;
typedef __attribute__((ext_vector_type(8))) float v8f;

constexpr int B  = 8;
constexpr int S  = 4096;
constexpr int DV = 1024;

constexpr int SCHUNKS = 8;               // S split into 8 chunks of 512 rows
constexpr int SCHUNK  = S / SCHUNKS;     // 512
constexpr int VTILES  = DV / 16;         // 64 tiles of 16 columns
constexpr int WAVES_PER_BLOCK = 8;       // 256 threads, wave32
constexpr int TOTAL_WAVES = B * VTILES * SCHUNKS;                 // 4096
constexpr int REDUCE_BLOCKS = TOTAL_WAVES / WAVES_PER_BLOCK;      // 512

// ---------------------------------------------------------------------------
// Phase 1: per-wave partial column sums via V_WMMA_F32_16X16X4_F32.
// A = all-ones 16x4 -> C[m,n] += sum_k B[k,n]; every C row holds the col sums.
// Each wave: (b, vtile of 16 cols, 512-row S chunk), 128 WMMAs.
// Writes 16 partials -> ws.partial[wave*16 + n]. Deterministic (no atomics).
// ---------------------------------------------------------------------------
__global__ __launch_bounds__(256) void colsum_wmma(const float* __restrict__ value,
                                                   float* __restrict__ partial) {
    const int lane = threadIdx.x & 31;
    const int wave = threadIdx.x >> 5;
    const int w    = blockIdx.x * WAVES_PER_BLOCK + wave;

    const int sc = w % SCHUNKS;
    const int vt = (w / SCHUNKS) % VTILES;
    const int b  = w / (SCHUNKS * VTILES);

    const int h = lane >> 4;      // half-wave: B VGPR0 holds K=2h, VGPR1 holds K=2h+1
    const int n = lane & 15;      // column within the 16-wide tile (N = lane % 16)

    const int s0 = sc * SCHUNK;
    const float* p = value + ((size_t)b * S + (size_t)(s0 + 2 * h)) * DV
                           + (size_t)vt * 16 + n;

    v8f acc = {};
    v2f ones;
    ones[0] = 1.0f;
    ones[1] = 1.0f;

#pragma unroll 4
    for (int it = 0; it < SCHUNK / 4; ++it) {
        v2f bm;
        bm[0] = p[0];        // row s + 2h
        bm[1] = p[DV];       // row s + 2h + 1
        p += 4 * (size_t)DV; // advance 4 rows (one K-step of the tile)
        // D = A(ones, 16x4) * B(4x16 tile of value) + C   -> column sums
        acc = __builtin_amdgcn_wmma_f32_16x16x4_f32(
            /*neg_a=*/false, ones, /*neg_b=*/false, bm,
            /*c_mod=*/(short)0, acc, /*reuse_a=*/false, /*reuse_b=*/false);
    }

    // C layout: VGPR0, lanes 0..15 = (M=0, N=lane); all rows identical here.
    if (lane < 16) {
        partial[(size_t)w * 16 + n] = acc[0];
    }
}

// ---------------------------------------------------------------------------
// Phase 2: fold the 8 S-chunk partials per (b,v) in fixed order.
// ---------------------------------------------------------------------------
__global__ __launch_bounds__(256) void reduce_partials(const float* __restrict__ partial,
                                                       float* __restrict__ colsum) {
    const int tid = blockIdx.x * blockDim.x + threadIdx.x; // 0 .. B*DV-1
    const int b  = tid / DV;
    const int v  = tid % DV;
    const int vt = v >> 4;
    const int n  = v & 15;

    const size_t base = (((size_t)b * VTILES + vt) * SCHUNKS) * 16 + n;
    float s = 0.0f;
#pragma unroll
    for (int sc = 0; sc < SCHUNKS; ++sc) {
        s += partial[base + (size_t)sc * 16];
    }
    colsum[tid] = s;
}

// ---------------------------------------------------------------------------
// Phase 3: broadcast colsum over S with 128-bit streaming stores.
// colsum (32 KB) is cache-resident; stores dominate -> HBM-bound.
// ---------------------------------------------------------------------------
__global__ __launch_bounds__(256) void broadcast_out(const v4f* __restrict__ colsum4,
                                                     v4f* __restrict__ out4) {
    const long long i = (long long)blockIdx.x * blockDim.x + threadIdx.x;
    const int v4 = (int)(i & (DV / 4 - 1));  // 0..255
    const int b  = (int)(i >> 20);           // i / (S * DV/4), 2^20 per batch
    out4[i] = colsum4[(size_t)b * (DV / 4) + v4];
}

// ---------------------------------------------------------------------------
extern "C" void kernel_launch(void* const* d_in, const int* in_sizes, int n_in,
                              void* d_out, int out_size, void* d_ws, size_t ws_size,
                              hipStream_t stream) {
    (void)in_sizes; (void)n_in; (void)out_size; (void)ws_size;
    // d_in[0] = query (provably unused: it cancels out of retrieved/denom)
    const float* value = (const float*)d_in[1];
    float* out = (float*)d_out;

    float* partial = (float*)d_ws;                         // TOTAL_WAVES*16 floats (256 KB)
    float* colsum  = partial + (size_t)TOTAL_WAVES * 16;   // B*DV floats (32 KB)

    colsum_wmma<<<REDUCE_BLOCKS, 256, 0, stream>>>(value, partial);

    reduce_partials<<<(B * DV) / 256, 256, 0, stream>>>(partial, colsum);

    const long long total4 = (long long)B * S * (DV / 4);  // 8,388,608 float4 stores
    broadcast_out<<<(int)(total4 / 256), 256, 0, stream>>>((const v4f*)colsum, (v4f*)out);
}